// ResidualGATBlock_7224134992235
// MI455X (gfx1250) — compile-verified
//
#include <hip/hip_runtime.h>
#include <math.h>

#define N_NODES 50000
#define N_EDGES 640000
#define D       128
#define NEG_SLOPE 0.2f
#define BN_EPS  1e-5f

// LDS row stride in floats after TDM padding (128 data + 16 pad dwords):
// makes the two WMMA K-group lane-halves hit disjoint LDS bank sets.
#define LDS_STRIDE 144

typedef float v2f __attribute__((ext_vector_type(2)));
typedef float v4f __attribute__((ext_vector_type(4)));
typedef float v8f __attribute__((ext_vector_type(8)));
typedef unsigned u32x4 __attribute__((ext_vector_type(4)));
typedef int i32x4 __attribute__((ext_vector_type(4)));
typedef int i32x8 __attribute__((ext_vector_type(8)));

// ---------------------------------------------------------------------------
// TDM: issue a 2-D f32 tile load (rows x cols, row-major, stride == cols)
// from global memory into LDS at byte offset ldsOff, adding 16 pad dwords
// after every 128 stored dwords (row stride in LDS = 144 dwords).
// Descriptor layout per CDNA5 ISA 8.3/8.4:
//   group0 = {count|flags, lds_addr, global_addr, type=2}
//   group1 = {data_size=4B | pad_enable | pad_interval=128dw | pad_amount=16dw,
//             tensor_dim0/1, tile_dim0/1, dim0_stride}
// Groups 2/3 zero (2-D tensor).  Tracked by TENSORcnt.
// ---------------------------------------------------------------------------
__device__ __forceinline__ void tdm_load_2d_f32(const void* gsrc, unsigned ldsOff,
                                                unsigned rows, unsigned cols) {
  unsigned long long ga = (unsigned long long)(size_t)gsrc;
  u32x4 g0 = { 1u,                                   // count=1 (valid), user mode
               ldsOff,                               // lds_addr (bytes)
               (unsigned)(ga & 0xFFFFFFFFu),         // global_addr[31:0]
               (unsigned)((ga >> 32) & 0x01FFFFFFu)  // global_addr[56:32]
                 | (2u << 30) };                     // type=2 ("image")
  i32x8 g1 = { (int)((2u << 16)      // wg_mask=0, data_size=2 -> 4 bytes
                   | (1u << 20)      // pad_enable
                   | (6u << 22)      // pad_interval code 6 -> 128 dwords
                   | (15u << 25)),   // pad_amount  code 15 -> 16 dwords
               (int)(cols << 16),    // [63:48] tensor_dim0 lo
               (int)(rows << 16),    // [95:80] tensor_dim1 lo (dim0 hi = 0)
               (int)(cols << 16),    // [127:112] tile_dim0   (dim1 hi = 0)
               (int)rows,            // [143:128] tile_dim1, tile_dim2=0
               (int)cols,            // tensor_dim0_stride[31:0]
               0, 0 };
  i32x4 g2 = { 0, 0, 0, 0 };
  i32x4 g3 = { 0, 0, 0, 0 };
#if __clang_major__ >= 23
  i32x8 g4 = { 0, 0, 0, 0, 0, 0, 0, 0 };
  __builtin_amdgcn_tensor_load_to_lds(g0, g1, g2, g3, g4, 0);
#else
  __builtin_amdgcn_tensor_load_to_lds(g0, g1, g2, g3, 0);
#endif
}

// ---------------------------------------------------------------------------
// K0: zero scratch accumulators
// ---------------------------------------------------------------------------
__global__ void k_init(float* __restrict__ accum, float* __restrict__ den,
                       unsigned* __restrict__ maxkey,
                       float* __restrict__ cs, float* __restrict__ cs2) {
  size_t i = (size_t)blockIdx.x * blockDim.x + threadIdx.x;
  size_t stride = (size_t)gridDim.x * blockDim.x;
  for (size_t t = i; t < (size_t)N_NODES * D; t += stride) accum[t] = 0.f;
  for (size_t t = i; t < (size_t)N_NODES; t += stride) { den[t] = 0.f; maxkey[t] = 0u; }
  if (i < D) { cs[i] = 0.f; cs2[i] = 0.f; }
}

// ---------------------------------------------------------------------------
// K1: xl = x@Wl + bl ; xr = x@Wr + br   via V_WMMA_F32_16X16X4_F32
// 8 waves/block; Wl+Wr staged once per block into (bank-pad-swizzled) LDS by
// the Tensor Data Mover, then each wave loops over 16-row tiles (persistent
// grid).  50000 rows = 3125 exact tiles.
//
// A (16x4 f32) lane layout: lanes 0-15 M=0..15, VGPR0/1 = K = 2*(lane>>4)+{0,1}
// B (4x16 f32): lane indexes N, VGPR+lane-group index K (symmetric to A)
// C/D (16x16 f32): lane = N (+16 for M>=8), VGPR r = M%8
// ---------------------------------------------------------------------------
#define GEMM_WAVES 8
#define GEMM_BLOCKS 98

__global__ void __launch_bounds__(256)
k_gemm(const float* __restrict__ x,
       const float* __restrict__ Wl, const float* __restrict__ bl,
       const float* __restrict__ Wr, const float* __restrict__ br,
       float* __restrict__ xl, float* __restrict__ xr) {
  // Sole static LDS object -> starts at LDS offset 0.
  // [0, 72K) = Wl (padded rows), [72K, 144K) = Wr (padded rows)
  __shared__ float sW[2 * LDS_STRIDE * D];

  const int wave = threadIdx.x >> 5;
  const int lane = threadIdx.x & 31;

  if (wave == 0) {
    tdm_load_2d_f32(Wl, 0u, D, D);
    tdm_load_2d_f32(Wr, (unsigned)(LDS_STRIDE * D * sizeof(float)), D, D);
    __builtin_amdgcn_s_wait_tensorcnt(0);
  }
  __syncthreads();
  // The TDM writes LDS through an address carried as descriptor *data*, which
  // LLVM's alias analysis cannot see.  Escape the LDS pointer into an opaque
  // asm with a memory clobber so the ds_loads below are not folded to undef.
  asm volatile("" : : "v"(&sW[0]) : "memory");

  const float* __restrict__ sWl = sW;
  const float* __restrict__ sWr = sW + LDS_STRIDE * D;

  const int mrow = lane & 15;              // M (for A) / N (for B,C)
  const int kg   = lane >> 4;              // K group

  for (int tile = blockIdx.x * GEMM_WAVES + wave; tile < N_NODES / 16;
       tile += GEMM_BLOCKS * GEMM_WAVES) {
    const int row0 = tile * 16;

    v8f accL[8], accR[8];
    const v8f vzero = {};
#pragma unroll
    for (int n = 0; n < 8; ++n) { accL[n] = vzero; accR[n] = vzero; }

    const float* __restrict__ xrow = x + (size_t)row0 * D;
    for (int k0 = 0; k0 < D; k0 += 4) {
      const int kb = k0 + kg * 2;
      const v2f a = *(const v2f*)(xrow + (size_t)mrow * D + kb);   // b64 load
#pragma unroll
      for (int n = 0; n < 8; ++n) {
        const int col = n * 16 + mrow;
        v2f bL, bR;
        bL.x = sWl[kb * LDS_STRIDE + col];
        bL.y = sWl[(kb + 1) * LDS_STRIDE + col];
        bR.x = sWr[kb * LDS_STRIDE + col];
        bR.y = sWr[(kb + 1) * LDS_STRIDE + col];
        accL[n] = __builtin_amdgcn_wmma_f32_16x16x4_f32(
            false, a, false, bL, (short)0, accL[n], false, false);
        accR[n] = __builtin_amdgcn_wmma_f32_16x16x4_f32(
            false, a, false, bR, (short)0, accR[n], false, false);
      }
    }

#pragma unroll
    for (int n = 0; n < 8; ++n) {
      const int col = n * 16 + mrow;
      const float bLv = bl[col];
      const float bRv = br[col];
#pragma unroll
      for (int r = 0; r < 8; ++r) {
        const int row = row0 + r + kg * 8;
        xl[(size_t)row * D + col] = accL[n][r] + bLv;
        xr[(size_t)row * D + col] = accR[n][r] + bRv;
      }
    }
  }
}

// monotonic uint key for float atomic max
__device__ __forceinline__ unsigned f2key(float f) {
  unsigned b = __float_as_uint(f);
  return b ^ ((b >> 31) ? 0xFFFFFFFFu : 0x80000000u);
}
__device__ __forceinline__ float key2f(unsigned k) {
  return __uint_as_float(k ^ ((k >> 31) ? 0x80000000u : 0xFFFFFFFFu));
}

// ---------------------------------------------------------------------------
// K2: per-edge logit = att . leaky_relu(xl[src] + xr[dst] + ea*We)
//     wave per edge, lane owns 4 features; segment-max via uint atomic max.
// ---------------------------------------------------------------------------
__global__ void __launch_bounds__(256)
k_edge_logits(const float* __restrict__ xl, const float* __restrict__ xr,
              const int* __restrict__ ei, const float* __restrict__ edge_attr,
              const float* __restrict__ We, const float* __restrict__ att,
              float* __restrict__ logits, unsigned* __restrict__ maxkey) {
  const int lane = threadIdx.x & 31;
  const int waveId = (int)((blockIdx.x * blockDim.x + threadIdx.x) >> 5);
  const int nWaves = (int)((gridDim.x * blockDim.x) >> 5);
  const int d0 = lane * 4;
  const v4f we = *(const v4f*)(We + d0);
  const v4f at = *(const v4f*)(att + d0);
  const int* __restrict__ src = ei;
  const int* __restrict__ dst = ei + N_EDGES;

  for (int e = waveId; e < N_EDGES; e += nWaves) {
    const int s = src[e], t = dst[e];
    const float ea = edge_attr[e];
    const v4f a = *(const v4f*)(xl + (size_t)s * D + d0);
    const v4f b = *(const v4f*)(xr + (size_t)t * D + d0);
    float p = 0.f;
#pragma unroll
    for (int j = 0; j < 4; ++j) {
      float m = a[j] + b[j] + ea * we[j];
      m = (m > 0.f) ? m : NEG_SLOPE * m;
      p += m * at[j];
    }
    for (int off = 16; off > 0; off >>= 1) p += __shfl_down(p, off, 32);
    if (lane == 0) {
      logits[e] = p;
      __hip_atomic_fetch_max(&maxkey[t], f2key(p),
                             __ATOMIC_RELAXED, __HIP_MEMORY_SCOPE_AGENT);
    }
  }
}

// ---------------------------------------------------------------------------
// K3: ex = exp(logit - max[dst]); den[dst] += ex (overwrite logits with ex)
// ---------------------------------------------------------------------------
__global__ void __launch_bounds__(256)
k_edge_exp(float* __restrict__ logits, const int* __restrict__ ei,
           const unsigned* __restrict__ maxkey, float* __restrict__ den) {
  const int e = blockIdx.x * blockDim.x + threadIdx.x;
  if (e >= N_EDGES) return;
  const int t = ei[N_EDGES + e];
  const float ex = expf(logits[e] - key2f(maxkey[t]));
  logits[e] = ex;
  __hip_atomic_fetch_add(&den[t], ex, __ATOMIC_RELAXED, __HIP_MEMORY_SCOPE_AGENT);
}

// ---------------------------------------------------------------------------
// K4: accum[dst] += xl[src] * (ex / (den[dst]+1e-16)); wave per edge
// ---------------------------------------------------------------------------
__global__ void __launch_bounds__(256)
k_edge_scatter(const float* __restrict__ xl, const float* __restrict__ ex,
               const float* __restrict__ den, const int* __restrict__ ei,
               float* __restrict__ accum) {
  const int lane = threadIdx.x & 31;
  const int waveId = (int)((blockIdx.x * blockDim.x + threadIdx.x) >> 5);
  const int nWaves = (int)((gridDim.x * blockDim.x) >> 5);
  const int d0 = lane * 4;
  const int* __restrict__ src = ei;
  const int* __restrict__ dst = ei + N_EDGES;

  for (int e = waveId; e < N_EDGES; e += nWaves) {
    if (e + nWaves < N_EDGES) {                    // global_prefetch_b8
      __builtin_prefetch(&src[e + nWaves], 0, 1);
      __builtin_prefetch(&dst[e + nWaves], 0, 1);
    }
    const int s = src[e], t = dst[e];
    const float alpha = ex[e] / (den[t] + 1e-16f);
    const float* __restrict__ xs = xl + (size_t)s * D + d0;
    float* __restrict__ ap = accum + (size_t)t * D + d0;
    const v4f v = *(const v4f*)xs;
#pragma unroll
    for (int j = 0; j < 4; ++j)
      __hip_atomic_fetch_add(&ap[j], v[j] * alpha,
                             __ATOMIC_RELAXED, __HIP_MEMORY_SCOPE_AGENT);
  }
}

// ---------------------------------------------------------------------------
// K5: h = gelu_erf(x + accum + bias_out); column partial sums for BN
// ---------------------------------------------------------------------------
#define ROWS_PER_BLOCK 256
__global__ void __launch_bounds__(128)
k_gelu_stats(const float* __restrict__ x, const float* __restrict__ bias_out,
             float* __restrict__ h, float* __restrict__ cs, float* __restrict__ cs2) {
  const int d = threadIdx.x;
  const int r0 = blockIdx.x * ROWS_PER_BLOCK;
  const int r1 = (r0 + ROWS_PER_BLOCK < N_NODES) ? r0 + ROWS_PER_BLOCK : N_NODES;
  const float bo = bias_out[d];
  float s = 0.f, s2 = 0.f;
  for (int r = r0; r < r1; ++r) {
    const size_t idx = (size_t)r * D + d;
    const float v = x[idx] + h[idx] + bo;
    const float g = 0.5f * v * (1.f + erff(v * 0.70710678118654752f));
    h[idx] = g;
    s += g;
    s2 += g * g;
  }
  __hip_atomic_fetch_add(&cs[d], s, __ATOMIC_RELAXED, __HIP_MEMORY_SCOPE_AGENT);
  __hip_atomic_fetch_add(&cs2[d], s2, __ATOMIC_RELAXED, __HIP_MEMORY_SCOPE_AGENT);
}

// ---------------------------------------------------------------------------
// K6: out = (h - mean) * rsqrt(var + eps) * gamma + beta
// ---------------------------------------------------------------------------
__global__ void __launch_bounds__(256)
k_bn(const float* __restrict__ h, const float* __restrict__ cs,
     const float* __restrict__ cs2, const float* __restrict__ gamma,
     const float* __restrict__ beta, float* __restrict__ out) {
  const size_t i = (size_t)blockIdx.x * blockDim.x + threadIdx.x;
  if (i >= (size_t)N_NODES * D) return;
  const int d = (int)(i & (D - 1));
  const float inv_n = 1.f / (float)N_NODES;
  const float mean = cs[d] * inv_n;
  const float var = cs2[d] * inv_n - mean * mean;
  out[i] = (h[i] - mean) * rsqrtf(var + BN_EPS) * gamma[d] + beta[d];
}

// ---------------------------------------------------------------------------
// host side
// ---------------------------------------------------------------------------
extern "C" void kernel_launch(void* const* d_in, const int* in_sizes, int n_in,
                              void* d_out, int out_size, void* d_ws, size_t ws_size,
                              hipStream_t stream) {
  const float* x        = (const float*)d_in[0];
  const int*   ei       = (const int*)  d_in[1];   // [2, E]
  const float* eattr    = (const float*)d_in[2];   // [E, 1]
  const float* Wl       = (const float*)d_in[3];
  const float* bl       = (const float*)d_in[4];
  const float* Wr       = (const float*)d_in[5];
  const float* br       = (const float*)d_in[6];
  const float* We       = (const float*)d_in[7];   // [1, D]
  const float* att      = (const float*)d_in[8];
  const float* bias_out = (const float*)d_in[9];
  const float* gamma    = (const float*)d_in[10];
  const float* beta     = (const float*)d_in[11];
  float* out = (float*)d_out;

  // workspace slices
  char* ws = (char*)d_ws;
  float*    xl     = (float*)ws;                       ws += (size_t)N_NODES * D * 4;
  float*    xr     = (float*)ws;                       ws += (size_t)N_NODES * D * 4;
  float*    accum  = (float*)ws;                       ws += (size_t)N_NODES * D * 4;
  float*    logits = (float*)ws;                       ws += (size_t)N_EDGES * 4;
  float*    den    = (float*)ws;                       ws += (size_t)N_NODES * 4;
  unsigned* maxkey = (unsigned*)ws;                    ws += (size_t)N_NODES * 4;
  float*    cs     = (float*)ws;                       ws += D * 4;
  float*    cs2    = (float*)ws;

  hipLaunchKernelGGL(k_init, dim3(2048), dim3(256), 0, stream,
                     accum, den, maxkey, cs, cs2);

  hipLaunchKernelGGL(k_gemm, dim3(GEMM_BLOCKS), dim3(256), 0, stream,
                     x, Wl, bl, Wr, br, xl, xr);

  hipLaunchKernelGGL(k_edge_logits, dim3(4096), dim3(256), 0, stream,
                     xl, xr, ei, eattr, We, att, logits, maxkey);

  hipLaunchKernelGGL(k_edge_exp, dim3((N_EDGES + 255) / 256), dim3(256), 0, stream,
                     logits, ei, maxkey, den);

  hipLaunchKernelGGL(k_edge_scatter, dim3(4096), dim3(256), 0, stream,
                     xl, logits, den, ei, accum);

  hipLaunchKernelGGL(k_gelu_stats,
                     dim3((N_NODES + ROWS_PER_BLOCK - 1) / ROWS_PER_BLOCK),
                     dim3(128), 0, stream,
                     x, bias_out, accum, cs, cs2);

  hipLaunchKernelGGL(k_bn, dim3(((size_t)N_NODES * D + 255) / 256), dim3(256), 0, stream,
                     accum, cs, cs2, gamma, beta, out);
}